// Sampler_28845000360542
// MI455X (gfx1250) — compile-verified
//
#include <hip/hip_runtime.h>
#include <hip/hip_bf16.h>
#include <stdint.h>

// ---------------- tunables / layout ----------------
#define NTHR    512          // 16 wave32s per row-workgroup
#define TILE    4096         // elements per TDM chunk (16 KB)
#define NBIN    4096         // 12-bit radix histogram
#define EQ_CAP  1024         // candidates inside the coarse threshold bin
#define FIN_CAP 320          // final candidate list (>= k + bin residue)
#define TOPP    0.9f
#define INV_T   1.25f        // 1 / 0.8

// Dynamic LDS layout (bytes). No static __shared__ -> dynamic LDS starts at
// offset 0, so these byte offsets are also the TDM lds_addr values.
#define OFF_BUF   0                    // 2 * TILE floats  (32 KB)
#define OFF_HIST  32768                // NBIN u32         (16 KB)
#define OFF_BSUM  49152                // NTHR u32         ( 2 KB)
#define OFF_EQV   51200                // EQ_CAP f32       ( 4 KB)
#define OFF_EQI   55296                // EQ_CAP i32       ( 4 KB)
#define OFF_FIV   59392                // FIN_CAP f32
#define OFF_FII   60672                // FIN_CAP i32
#define OFF_SC    61952                // 16 ints of scalars
#define SMEM_BYTES 62016

typedef unsigned int u32x4 __attribute__((ext_vector_type(4)));
typedef int          i32x8 __attribute__((ext_vector_type(8)));
typedef int          i32x4 __attribute__((ext_vector_type(4)));

#if defined(__gfx1250__) && __has_builtin(__builtin_amdgcn_tensor_load_to_lds) && \
    __has_builtin(__builtin_amdgcn_s_wait_tensorcnt)
#define HAVE_TDM 1
#else
#define HAVE_TDM 0
#endif

static __device__ __forceinline__ int imin(int a, int b) { return a < b ? a : b; }
static __device__ __forceinline__ int imax(int a, int b) { return a > b ? a : b; }

// Order-preserving uint transform: larger float -> larger uint.
static __device__ __forceinline__ uint32_t ordered_bits(float f) {
  uint32_t x = __float_as_uint(f);
  return x ^ ((x & 0x80000000u) ? 0xFFFFFFFFu : 0x80000000u);
}

static __device__ __forceinline__ uint32_t rotl32(uint32_t x, int r) {
  return (x << r) | (x >> (32 - r));
}

// Threefry-2x32, 20 rounds — JAX's default PRNG.
static __device__ void threefry2x32(uint32_t k0, uint32_t k1, uint32_t x0, uint32_t x1,
                                    uint32_t& o0, uint32_t& o1) {
  uint32_t ks[3] = {k0, k1, 0x1BD11BDAu ^ k0 ^ k1};
  const int R0[4] = {13, 15, 26, 6};
  const int R1[4] = {17, 29, 16, 24};
  x0 += ks[0]; x1 += ks[1];
  for (int g = 0; g < 5; ++g) {
    const int* R = (g & 1) ? R1 : R0;
    for (int i = 0; i < 4; ++i) { x0 += x1; x1 = rotl32(x1, R[i]); x1 ^= x0; }
    x0 += ks[(g + 1) % 3];
    x1 += ks[(g + 2) % 3] + (uint32_t)(g + 1);
  }
  o0 = x0; o1 = x1;
}

// Issue one TDM load: `tile_elems` f32 from gsrc into LDS byte offset lds_off.
// tensor_dim0 = n_valid so out-of-range tail elements zero-fill.
// 6-arg builtin form (clang-23 / therock-10.0): (u32x4, i32x8, i32x4, i32x4, i32x8, i32).
static __device__ __forceinline__ void tdm_issue(const float* gsrc, uint32_t lds_off,
                                                 uint32_t n_valid, uint32_t tile_elems) {
#if HAVE_TDM
  uint64_t ga = (uint64_t)(uintptr_t)gsrc;
  u32x4 g0;
  g0[0] = 1u;                                                  // count=1, user mode
  g0[1] = lds_off;                                             // lds_addr (bytes)
  g0[2] = (uint32_t)(ga & 0xFFFFFFFFu);                        // global_addr[31:0]
  g0[3] = (uint32_t)((ga >> 32) & 0x01FFFFFFu) | (2u << 30);   // addr[56:32] | type=2
  i32x8 g1;
  g1[0] = 0x00020000;                                          // data_size=2 (4B)
  g1[1] = (int)((n_valid & 0xFFFFu) << 16);                    // tensor_dim0 lo
  g1[2] = (int)(((n_valid >> 16) & 0xFFFFu) | (1u << 16));     // dim0 hi | tensor_dim1=1
  g1[3] = (int)((tile_elems & 0xFFFFu) << 16);                 // tile_dim0
  g1[4] = 1;                                                   // tile_dim1=1, tile_dim2=0
  g1[5] = (int)tile_elems;                                     // tensor_dim0_stride (unused)
  g1[6] = 0;
  g1[7] = 0;
  i32x4 z4 = {0, 0, 0, 0};
  i32x8 z8 = {0, 0, 0, 0, 0, 0, 0, 0};
  __builtin_amdgcn_tensor_load_to_lds(g0, g1, z4, z4, z8, 0);
#else
  (void)gsrc; (void)lds_off; (void)n_valid; (void)tile_elems;
#endif
}

// Find largest bin b with suffix-count(b) >= K over NBIN-entry histogram.
// All NTHR threads participate. Writes *outBin=b and *outAbove=suffix(b+1).
static __device__ void find_kth_bin(uint32_t* hist, uint32_t* bsum, uint32_t K,
                                    int* outBin, int* outAbove) {
  const int t = threadIdx.x;
  uint32_t s = 0;
  for (int j = 0; j < NBIN / NTHR; ++j) s += hist[t * (NBIN / NTHR) + j];
  bsum[t] = s;
  __syncthreads();
  for (int step = 1; step < NTHR; step <<= 1) {   // suffix Hillis-Steele scan
    uint32_t v = bsum[t] + ((t + step < NTHR) ? bsum[t + step] : 0u);
    __syncthreads();
    bsum[t] = v;
    __syncthreads();
  }
  uint32_t running = (t + 1 < NTHR) ? bsum[t + 1] : 0u;
  for (int j = NBIN / NTHR - 1; j >= 0; --j) {
    int b = t * (NBIN / NTHR) + j;
    uint32_t nr = running + hist[b];
    if (running < K && nr >= K) { *outBin = b; *outAbove = (int)running; }
    running = nr;
  }
  __syncthreads();
}

__global__ __launch_bounds__(NTHR)
void sampler_topk_topp_kernel(const float* __restrict__ logits,
                              const int* __restrict__ kptr,
                              int* __restrict__ out, int V) {
  extern __shared__ unsigned char smem[];
  float*    buf  = (float*)(smem + OFF_BUF);
  uint32_t* hist = (uint32_t*)(smem + OFF_HIST);
  uint32_t* bsum = (uint32_t*)(smem + OFF_BSUM);
  float*    eqV  = (float*)(smem + OFF_EQV);
  int*      eqI  = (int*)(smem + OFF_EQI);
  float*    fiV  = (float*)(smem + OFF_FIV);
  int*      fiI  = (int*)(smem + OFF_FII);
  int*      sc   = (int*)(smem + OFF_SC);
  // sc[0]=cntEq sc[1]=cntFin sc[2]=coarseBin sc[3]=aboveCoarse
  // sc[4]=refBin sc[5]=aboveRef sc[6]=answer sc[7]=m  sc[8]=rW(float bits)

  const int tid = threadIdx.x;
  const int row = blockIdx.x;
  const uint32_t K = (uint32_t)imax(1, imin(*kptr, FIN_CAP / 2));
  const float* rowp = logits + (size_t)row * (size_t)V;

  for (int i = tid; i < NBIN; i += NTHR) hist[i] = 0;
  if (tid < 16) sc[tid] = 0;
  __syncthreads();

  // ---- Pass 1: coarse 12-bit histogram, TDM double-buffered through LDS ----
  const int chunks = (V + TILE - 1) / TILE;
  const bool wave0 = (tid < 32);
#if HAVE_TDM
  if (wave0) {
    tdm_issue(rowp, OFF_BUF, (uint32_t)V, TILE);
    if (chunks > 1) tdm_issue(rowp + TILE, OFF_BUF + TILE * 4, (uint32_t)(V - TILE), TILE);
  }
  for (int c = 0; c < chunks; ++c) {
    if (wave0) {
      if (c + 2 <= chunks) __builtin_amdgcn_s_wait_tensorcnt(1);  // oldest of 2 done
      else                 __builtin_amdgcn_s_wait_tensorcnt(0);
    }
    __syncthreads();
    const float4* b4 = (const float4*)(buf + (c & 1) * TILE);
#pragma unroll
    for (int q = 0; q < TILE / (NTHR * 4); ++q) {
      float4 v = b4[tid * (TILE / (NTHR * 4)) + q];
      atomicAdd(&hist[ordered_bits(v.x) >> 20], 1u);
      atomicAdd(&hist[ordered_bits(v.y) >> 20], 1u);
      atomicAdd(&hist[ordered_bits(v.z) >> 20], 1u);
      atomicAdd(&hist[ordered_bits(v.w) >> 20], 1u);
    }
    __syncthreads();
    if (wave0 && (c + 2) < chunks) {
      tdm_issue(rowp + (size_t)(c + 2) * TILE, OFF_BUF + (c & 1) * TILE * 4,
                (uint32_t)(V - (c + 2) * TILE), TILE);
    }
  }
#else
  for (int i = tid; i < V; i += NTHR)
    atomicAdd(&hist[ordered_bits(rowp[i]) >> 20], 1u);
  __syncthreads();
#endif

  // ---- coarse threshold bin ----
  find_kth_bin(hist, bsum, K, &sc[2], &sc[3]);
  const int binB  = sc[2];
  const int above = sc[3];                 // strictly above the bin, < K
  for (int i = tid; i < NBIN; i += NTHR) hist[i] = 0;   // reuse for refinement
  __syncthreads();

  // ---- Pass 2 (L2-resident): route candidates, refine mid-12-bit histogram ----
  for (int i = tid; i < V; i += NTHR) {
    float v = rowp[i];
    uint32_t u = ordered_bits(v);
    int b = (int)(u >> 20);
    if (b > binB) {
      int p = atomicAdd(&sc[1], 1);
      if (p < FIN_CAP) { fiV[p] = v; fiI[p] = i; }
    } else if (b == binB) {
      atomicAdd(&hist[(u >> 8) & 0xFFFu], 1u);
      int p = atomicAdd(&sc[0], 1);
      if (p < EQ_CAP) { eqV[p] = v; eqI[p] = i; }
    }
  }
  __syncthreads();

  // ---- refined threshold within the coarse bin ----
  find_kth_bin(hist, bsum, K - (uint32_t)above, &sc[4], &sc[5]);
  const int b2 = sc[4];

  const int nEq = imin(sc[0], EQ_CAP);
  for (int i = tid; i < nEq; i += NTHR) {
    uint32_t u = ordered_bits(eqV[i]);
    if ((int)((u >> 8) & 0xFFFu) >= b2) {
      int p = atomicAdd(&sc[1], 1);
      if (p < FIN_CAP) { fiV[p] = eqV[i]; fiI[p] = eqI[i]; }
    }
  }
  __syncthreads();

  // ---- sort final candidates descending (odd-even transposition; ~k entries) ----
  const int F  = imin(sc[1], FIN_CAP);
  const int Fp = (F + 1) & ~1;
  if (tid == 0 && F < Fp) { fiV[F] = __uint_as_float(0xFF800000u); fiI[F] = V; }
  __syncthreads();
  for (int pass = 0; pass < Fp; ++pass) {
    int j = tid * 2 + (pass & 1);
    if (j + 1 < Fp) {
      float a = fiV[j], b = fiV[j + 1];
      if (a < b) {
        fiV[j] = b; fiV[j + 1] = a;
        int t2 = fiI[j]; fiI[j] = fiI[j + 1]; fiI[j + 1] = t2;
      }
    }
    __syncthreads();
  }

  // ---- softmax, top-p cut, Threefry RNG (thread 0; O(k)) ----
  float* wV = (float*)bsum;   // reuse
  if (tid == 0) {
    int kk = imin(F, (int)K);
    float x0 = fiV[0] * INV_T;
    float W = 0.f;
    for (int i = 0; i < kk; ++i) { float w = expf(fiV[i] * INV_T - x0); wV[i] = w; W += w; }
    int m = 1; float c = wV[0];                 // keep i iff i==0 or cdf[i-1] <= p
    for (int i = 1; i < kk; ++i) { if (c <= TOPP * W) { m = i + 1; c += wV[i]; } else break; }
    float Wm = 0.f;
    for (int i = 0; i < m; ++i) Wm += wV[i];
    // r = uniform(fold_in(key(0),1))[row], JAX threefry counter layout
    uint32_t k0, k1, o0, o1;
    threefry2x32(0u, 0u, 0u, 1u, k0, k1);       // fold_in(key(0), 1)
    uint32_t lo = (row < 128) ? (uint32_t)row : (uint32_t)(row - 128);
    threefry2x32(k0, k1, lo, lo + 128u, o0, o1);
    uint32_t bits = (row < 128) ? o0 : o1;
    float r = __uint_as_float(0x3F800000u | (bits >> 9)) - 1.0f;
    sc[6] = V;                                   // answer init (r past total mass -> V)
    sc[7] = m;
    sc[8] = (int)__float_as_uint(r * Wm);        // compare S_t > r*Wm (unnormalized)
  }
  __syncthreads();

  // ---- vocab-order inverse CDF: answer = min idx among kept with prefix > r ----
  const int   m  = sc[7];
  const float rW = __uint_as_float((uint32_t)sc[8]);
  if (tid < m) {
    int myI = fiI[tid];
    float S = 0.f;
    for (int i = 0; i < m; ++i) S += (fiI[i] <= myI) ? wV[i] : 0.f;
    if (S > rW) atomicMin(&sc[6], myI);
  }
  __syncthreads();
  if (tid == 0) out[row] = sc[6];
}

extern "C" void kernel_launch(void* const* d_in, const int* in_sizes, int n_in,
                              void* d_out, int out_size, void* d_ws, size_t ws_size,
                              hipStream_t stream) {
  const float* logits = (const float*)d_in[0];
  const int*   kptr   = (const int*)d_in[1];
  const int B = out_size;                 // 256 rows, one sampled id each
  const int V = in_sizes[0] / B;          // 50257
  sampler_topk_topp_kernel<<<B, NTHR, SMEM_BYTES, stream>>>(logits, kptr, (int*)d_out, V);
}